// EquiSchNet_1451698946132
// MI455X (gfx1250) — compile-verified
//
#include <hip/hip_runtime.h>
#include <math.h>
#include <cstdint>

// ---------------- problem constants ----------------
#define BS   16
#define A1   256
#define A2   256
#define AT   512
#define NE   24576
#define HH   128
#define FF   128
#define LL   6
#define NG   50
#define NGP  52            // NG padded to multiple of 4 for K-stepping
#define NC   4
#define ROWS (BS*AT)       // 8192 node rows
#define PI_F 3.14159265358979323846f

#define USE_ASYNC_LDS 1    // CDNA5 GLOBAL_LOAD_ASYNC_TO_LDS path (ASYNCcnt-tracked)

typedef float v2f __attribute__((ext_vector_type(2)));
typedef float v8f __attribute__((ext_vector_type(8)));

// D = A(16x4,f32) * B(4x16,f32) + C(16x16,f32) ; full-precision CDNA5 WMMA
__device__ __forceinline__ v8f wmma_k4(v2f a, v2f b, v8f c) {
    return __builtin_amdgcn_wmma_f32_16x16x4_f32(
        /*neg_a=*/false, a, /*neg_b=*/false, b,
        /*c_mod=*/(short)0, c, /*reuse_a=*/false, /*reuse_b=*/false);
}

// shifted softplus: log(1+exp(x)) - log(2)
__device__ __forceinline__ float ssp_f(float x) {
    float sp = (x > 20.0f) ? x : log1pf(expf(x));
    return sp - 0.6931471805599453f;
}

// Stage an 8KB (16x128 f32) tile global->LDS.
// Async path: per-lane CDNA5 async copy, no VGPR data movement; INST_OFFSET is
// added to BOTH the global and LDS addresses, so one base pair covers both halves.
// Low 32 bits of a generic __shared__ pointer are the LDS byte address (flat LDS
// aperture truncation, ISA 10.2), so integer operands are safe here.
__device__ __forceinline__ void stage_tile_8kb(const float* __restrict__ gsrc,
                                               float* lds_dst, int tid) {
#if USE_ASYNC_LDS
    uint32_t l0 = (uint32_t)(uintptr_t)lds_dst + (uint32_t)tid * 16u;
    uint64_t g0 = (uint64_t)(uintptr_t)gsrc + (uint64_t)tid * 16u;
    asm volatile("global_load_async_to_lds_b128 %0, %1, off"
                 :: "v"(l0), "v"(g0) : "memory");
    asm volatile("global_load_async_to_lds_b128 %0, %1, off offset:4096"
                 :: "v"(l0), "v"(g0) : "memory");
    asm volatile("s_wait_asynccnt 0" ::: "memory");
#else
    for (int i = tid; i < 16 * HH; i += 256) lds_dst[i] = gsrc[i];
#endif
}

// ---------------- 1) embedding: h = concat(sites@emb1, sites_p@emb2) ----------------
__global__ void emb_kernel(const float* __restrict__ sites,
                           const float* __restrict__ sites_p,
                           const float* __restrict__ e1W, const float* __restrict__ e1b,
                           const float* __restrict__ e2W, const float* __restrict__ e2b,
                           float* __restrict__ h) {
    int idx  = blockIdx.x * 256 + threadIdx.x;       // BS*AT*HH threads
    int j    = idx & (HH - 1);
    int node = idx >> 7;
    int a    = node & (AT - 1);
    int b    = node >> 9;
    float v;
    if (a < A1) {
        v = sites[b * A1 + a] * e1W[j] + e1b[j];
    } else {
        int a2 = a - A1;
        const float* sp = sites_p + (b * A2 + a2) * 2;
        v = sp[0] * e2W[j] + sp[1] * e2W[HH + j] + e2b[j];
    }
    h[idx] = v;
}

// ---------------- 2) edge features: RBF (zero-padded to NGP) + cosine cutoff ----------------
__global__ void edge_pre_kernel(const float* __restrict__ ew,
                                float* __restrict__ eattr,   // NE x NGP
                                float* __restrict__ ccut) {  // NE
    int idx = blockIdx.x * 256 + threadIdx.x;        // NE*NGP threads
    int e = idx / NGP, g = idx % NGP;
    float w = ew[e];
    float val = 0.0f;
    if (g < NG) {
        const float step  = 10.0f / 49.0f;
        const float coeff = -0.5f / (step * step);
        float d = w - (float)g * step;
        val = expf(coeff * d * d);
    }
    eattr[idx] = val;
    if (g == 0) ccut[e] = 0.5f * (cosf(w * PI_F * 0.1f) + 1.0f);
}

// ---------------- 3) edge filter MLP (per layer) ----------------
// 4 waves/block: wave c computes color c for a 16-edge tile, then the
// edge's own color is selected at writeout (mirrors reference gather).
__global__ __launch_bounds__(128)
void edge_mlp_kernel(const float* __restrict__ eattr, const float* __restrict__ ccut,
                     const int*   __restrict__ colors,
                     const float* __restrict__ W1, const float* __restrict__ b1,  // NCx50x128 / NCx128
                     const float* __restrict__ W2, const float* __restrict__ b2,  // NCx128x128 / NCx128
                     float* __restrict__ wedge) {                                 // NE x 128
    __shared__ __align__(16) float sA[16 * NGP];
    __shared__ __align__(16) float sT[NC][16 * FF];
    const int tid  = threadIdx.x;
    const int c    = tid >> 5;           // wave == color
    const int lane = tid & 31;
    const int hi   = lane >> 4, lo = lane & 15;
    const int e0   = blockIdx.x * 16;

    for (int i = tid; i < 16 * NGP; i += 128) sA[i] = eattr[e0 * NGP + i];
    __syncthreads();

    const float* W1c = W1 + c * NG * FF;
    const float* b1c = b1 + c * FF;
    const float* W2c = W2 + c * FF * FF;
    const float* b2c = b2 + c * FF;
    float* sTc = &sT[c][0];

    // GEMM1: (16 x 50) @ (50 x 128), +bias, ssp -> sTc
    for (int q = 0; q < 8; ++q) {
        const int n = q * 16 + lo;
        v8f acc = {};
        for (int k0 = 0; k0 < NGP; k0 += 4) {
            const int kk = k0 + 2 * hi;
            v2f a, b;
            a.x = sA[lo * NGP + kk];
            a.y = sA[lo * NGP + kk + 1];
            b.x = (kk     < NG) ? W1c[kk * FF + n]       : 0.0f;
            b.y = (kk + 1 < NG) ? W1c[(kk + 1) * FF + n] : 0.0f;
            acc = wmma_k4(a, b, acc);
        }
        const float bias = b1c[n];
        #pragma unroll
        for (int i = 0; i < 8; ++i) {
            const int row = i + 8 * hi;
            sTc[row * FF + n] = ssp_f(acc[i] + bias);
        }
    }
    __syncthreads();

    // GEMM2: (16 x 128) @ (128 x 128), +bias, per-edge color select, * Ccut
    for (int q = 0; q < 8; ++q) {
        const int n = q * 16 + lo;
        v8f acc = {};
        for (int k0 = 0; k0 < FF; k0 += 4) {
            const int kk = k0 + 2 * hi;
            v2f a, b;
            a.x = sTc[lo * FF + kk];
            a.y = sTc[lo * FF + kk + 1];
            b.x = W2c[kk * FF + n];
            b.y = W2c[(kk + 1) * FF + n];
            acc = wmma_k4(a, b, acc);
        }
        const float bias = b2c[n];
        #pragma unroll
        for (int i = 0; i < 8; ++i) {
            const int row = i + 8 * hi;
            const int e   = e0 + row;
            if (colors[e] == c)
                wedge[e * FF + n] = (acc[i] + bias) * ccut[e];
        }
    }
}

// ---------------- 4) xf = h @ conv1 (8192x128 @ 128x128) ----------------
__global__ __launch_bounds__(256)
void gemm_xf_kernel(const float* __restrict__ A, const float* __restrict__ B,
                    float* __restrict__ C) {
    __shared__ __align__(16) float sA[16 * HH];
    const int tid  = threadIdx.x;
    const int lane = tid & 31;
    const int hi   = lane >> 4, lo = lane & 15;
    const int r0   = blockIdx.x * 16;
    stage_tile_8kb(A + r0 * HH, sA, tid);
    __syncthreads();

    const int n = (tid >> 5) * 16 + lo;   // wave -> column tile
    v8f acc = {};
    for (int k0 = 0; k0 < HH; k0 += 4) {
        const int kk = k0 + 2 * hi;
        if (k0 + 8 < HH) __builtin_prefetch(&B[(k0 + 8) * HH + n], 0, 0);
        v2f a, b;
        a.x = sA[lo * HH + kk];
        a.y = sA[lo * HH + kk + 1];
        b.x = B[kk * HH + n];
        b.y = B[(kk + 1) * HH + n];
        acc = wmma_k4(a, b, acc);
    }
    #pragma unroll
    for (int i = 0; i < 8; ++i) {
        const int row = i + 8 * hi;
        C[(r0 + row) * HH + n] = acc[i];
    }
}

// ---------------- 5) scatter: agg[b,dst] += xf[b,src] * W[e] ----------------
__global__ void zero_kernel(float* __restrict__ p, int n) {
    int i = blockIdx.x * 256 + threadIdx.x;
    if (i < n) p[i] = 0.0f;
}

__global__ void scatter_kernel(const float* __restrict__ xf, const float* __restrict__ wedge,
                               const int* __restrict__ eidx, float* __restrict__ agg) {
    const int eb = blockIdx.x;            // NE*BS blocks
    const int e  = eb >> 4;
    const int b  = eb & (BS - 1);
    const int t  = threadIdx.x;           // feature 0..127
    const int src = eidx[2 * e];
    const int dst = eidx[2 * e + 1];
    const float m = xf[(b * AT + src) * HH + t] * wedge[e * HH + t];
    atomicAdd(&agg[(b * AT + dst) * HH + t], m);
}

// ---------------- 6) fused: h += blk( ssp(agg @ conv2 + b2) ) ----------------
__global__ __launch_bounds__(256)
void conv2_blk_kernel(const float* __restrict__ agg,
                      const float* __restrict__ W2, const float* __restrict__ b2v,
                      const float* __restrict__ Wb, const float* __restrict__ bbv,
                      float* __restrict__ h) {
    __shared__ __align__(16) float sA[16 * HH];
    __shared__ __align__(16) float sY[16 * HH];
    const int tid  = threadIdx.x;
    const int lane = tid & 31;
    const int hi   = lane >> 4, lo = lane & 15;
    const int r0   = blockIdx.x * 16;
    stage_tile_8kb(agg + r0 * HH, sA, tid);
    __syncthreads();

    const int n = (tid >> 5) * 16 + lo;
    // y = ssp(agg @ conv2 + b2)
    {
        v8f acc = {};
        for (int k0 = 0; k0 < HH; k0 += 4) {
            const int kk = k0 + 2 * hi;
            v2f a, b;
            a.x = sA[lo * HH + kk];
            a.y = sA[lo * HH + kk + 1];
            b.x = W2[kk * HH + n];
            b.y = W2[(kk + 1) * HH + n];
            acc = wmma_k4(a, b, acc);
        }
        const float bias = b2v[n];
        #pragma unroll
        for (int i = 0; i < 8; ++i) {
            const int row = i + 8 * hi;
            sY[row * HH + n] = ssp_f(acc[i] + bias);
        }
    }
    __syncthreads();
    // h += y @ blk + bb
    {
        v8f acc = {};
        for (int k0 = 0; k0 < HH; k0 += 4) {
            const int kk = k0 + 2 * hi;
            v2f a, b;
            a.x = sY[lo * HH + kk];
            a.y = sY[lo * HH + kk + 1];
            b.x = Wb[kk * HH + n];
            b.y = Wb[(kk + 1) * HH + n];
            acc = wmma_k4(a, b, acc);
        }
        const float bias = bbv[n];
        #pragma unroll
        for (int i = 0; i < 8; ++i) {
            const int row = i + 8 * hi;
            const int o   = (r0 + row) * HH + n;
            h[o] = h[o] + acc[i] + bias;
        }
    }
}

// ---------------- 7) folded linear head ----------------
// out[b] = sum_a h[b,a,:].dot(w) + AT*c ;  w = out1_W@out2_W, c = out1_b.out2_W + out2_b
__global__ void wvec_kernel(const float* __restrict__ o1W, const float* __restrict__ o1b,
                            const float* __restrict__ o2W, const float* __restrict__ o2b,
                            float* __restrict__ wv) {
    int i = threadIdx.x;                  // 128
    float s = 0.0f;
    for (int j = 0; j < HH / 2; ++j) s += o1W[i * (HH / 2) + j] * o2W[j];
    wv[i] = s;
    if (i == 0) {
        float c = o2b[0];
        for (int j = 0; j < HH / 2; ++j) c += o1b[j] * o2W[j];
        wv[HH] = c;
    }
}

__global__ void reduce_kernel(const float* __restrict__ h, const float* __restrict__ wv,
                              float* __restrict__ out) {
    __shared__ float red[256];
    const int b = blockIdx.x, tid = threadIdx.x;
    const float* hb = h + b * AT * HH;
    float s = 0.0f;
    for (int i = tid; i < AT * HH; i += 256) s += hb[i] * wv[i & (HH - 1)];
    red[tid] = s;
    __syncthreads();
    for (int st = 128; st > 0; st >>= 1) {
        if (tid < st) red[tid] += red[tid + st];
        __syncthreads();
    }
    if (tid == 0) out[b] = red[0] + (float)AT * wv[HH];
}

// ---------------- host launcher ----------------
extern "C" void kernel_launch(void* const* d_in, const int* in_sizes, int n_in,
                              void* d_out, int out_size, void* d_ws, size_t ws_size,
                              hipStream_t stream) {
    const float* sites   = (const float*)d_in[0];
    const float* sites_p = (const float*)d_in[1];
    const int*   eidx    = (const int*)  d_in[2];
    const float* ew      = (const float*)d_in[3];
    const int*   colors  = (const int*)  d_in[4];
    const float* e1W = (const float*)d_in[5];
    const float* e1b = (const float*)d_in[6];
    const float* e2W = (const float*)d_in[7];
    const float* e2b = (const float*)d_in[8];
    const float* conv1W = (const float*)d_in[9];
    const float* conv2W = (const float*)d_in[10];
    const float* conv2b = (const float*)d_in[11];
    const float* mlpW1 = (const float*)d_in[12];
    const float* mlpb1 = (const float*)d_in[13];
    const float* mlpW2 = (const float*)d_in[14];
    const float* mlpb2 = (const float*)d_in[15];
    const float* blkW  = (const float*)d_in[16];
    const float* blkb  = (const float*)d_in[17];
    const float* o1W = (const float*)d_in[18];
    const float* o1b = (const float*)d_in[19];
    const float* o2W = (const float*)d_in[20];
    const float* o2b = (const float*)d_in[21];

    float* ws    = (float*)d_ws;
    float* h     = ws;                        // ROWS*HH      = 1,048,576
    float* xf    = h     + ROWS * HH;         // ROWS*HH
    float* agg   = xf    + ROWS * HH;         // ROWS*HH
    float* eattr = agg   + ROWS * HH;         // NE*NGP       = 1,277,952
    float* ccut  = eattr + NE * NGP;          // NE
    float* wedge = ccut  + NE;                // NE*HH        = 3,145,728
    float* wv    = wedge + NE * HH;           // HH+1

    emb_kernel     <<<(ROWS * HH) / 256, 256, 0, stream>>>(sites, sites_p, e1W, e1b, e2W, e2b, h);
    edge_pre_kernel<<<(NE * NGP) / 256,  256, 0, stream>>>(ew, eattr, ccut);
    wvec_kernel    <<<1, HH, 0, stream>>>(o1W, o1b, o2W, o2b, wv);

    for (int l = 0; l < LL; ++l) {
        edge_mlp_kernel<<<NE / 16, 128, 0, stream>>>(
            eattr, ccut, colors,
            mlpW1 + l * NC * NG * FF, mlpb1 + l * NC * FF,
            mlpW2 + l * NC * FF * FF, mlpb2 + l * NC * FF, wedge);
        gemm_xf_kernel<<<ROWS / 16, 256, 0, stream>>>(h, conv1W + l * HH * FF, xf);
        zero_kernel<<<(ROWS * HH) / 256, 256, 0, stream>>>(agg, ROWS * HH);
        scatter_kernel<<<NE * BS, HH, 0, stream>>>(xf, wedge, eidx, agg);
        conv2_blk_kernel<<<ROWS / 16, 256, 0, stream>>>(
            agg, conv2W + l * FF * HH, conv2b + l * HH,
            blkW + l * HH * HH, blkb + l * HH, h);
    }
    reduce_kernel<<<BS, 256, 0, stream>>>(h, wv, (float*)d_out);
}